// GCNRecommendationModel_26852135535044
// MI455X (gfx1250) — compile-verified
//
#include <hip/hip_runtime.h>
#include <hip/hip_bf16.h>

// ---------------------------------------------------------------------------
// GCN: 3x (GCNConv -> BN -> ReLU) -> Linear(128,1)
//   N=100000 nodes, E=800000 edges, DIN=256, H=128, fp32 throughout.
// GEMMs use V_WMMA_F32_16X16X4_F32 (full fp32 precision, CDNA5 matrix path).
// Aggregation = self-loop init + per-edge wave scatter with f32 atomics; the
// 51 MB feature matrix is L2-resident (192 MB L2), so random traffic stays
// on-chip.
// ---------------------------------------------------------------------------

typedef __attribute__((ext_vector_type(2))) float v2f;
typedef __attribute__((ext_vector_type(8))) float v8f;

#define GCN_N   100000
#define GCN_E   800000
#define GCN_DIN 256
#define GCN_H   128

// ---------------- small utility kernels ----------------

__global__ void k_fill(float* __restrict__ p, float v, int n) {
    int i = blockIdx.x * blockDim.x + threadIdx.x;
    if (i < n) p[i] = v;
}

__global__ void k_deg(const int* __restrict__ dst, float* __restrict__ deg, int e) {
    int i = blockIdx.x * blockDim.x + threadIdx.x;
    if (i < e) atomicAdd(&deg[dst[i]], 1.0f);
}

__global__ void k_rsqrt(float* __restrict__ p, int n) {
    int i = blockIdx.x * blockDim.x + threadIdx.x;
    if (i < n) p[i] = rsqrtf(p[i]);
}

// ---------------- WMMA fp32 GEMM: C[n,128] = A[n,K] @ W[K,128] ----------------
// One wave -> one 16-row strip, all 128 output columns (8 accumulators).
__global__ void __launch_bounds__(256)
k_gemm_wmma(const float* __restrict__ A, const float* __restrict__ W,
            float* __restrict__ C, int nRows, int K) {
    const int wave    = (int)((blockIdx.x * blockDim.x + threadIdx.x) >> 5);
    const int rowBase = wave * 16;
    if (rowBase >= nRows) return;              // uniform per wave: EXEC stays all-1s

    const int lane  = threadIdx.x & 31;
    const int lhalf = lane >> 4;               // 0 = lanes 0-15, 1 = lanes 16-31
    const int llow  = lane & 15;

    const float* __restrict__ Arow = A + (size_t)(rowBase + llow) * K;

    v8f acc[8] = {};                           // zero-init accumulators

    for (int kb = 0; kb < K; kb += 4) {
        // A fragment (16x4 f32): VGPR0 = K=kb (lo lanes)/kb+2 (hi lanes), VGPR1 = kb+1/kb+3
        float2 av = *(const float2*)(Arow + kb + 2 * lhalf);
        v2f a; a.x = av.x; a.y = av.y;

        const float* __restrict__ Wk = W + (size_t)(kb + 2 * lhalf) * GCN_H + llow;
#pragma unroll
        for (int t = 0; t < 8; ++t) {
            // B fragment (4x16 f32): row-striped across lanes within a VGPR
            v2f b;
            b.x = Wk[t * 16];                  // row kb + 2*lhalf
            b.y = Wk[t * 16 + GCN_H];          // row kb + 2*lhalf + 1
            acc[t] = __builtin_amdgcn_wmma_f32_16x16x4_f32(
                false, a, false, b, (short)0, acc[t], false, false);
        }
    }

    // C/D layout: VGPR v, lanes 0-15 -> row v, lanes 16-31 -> row v+8
#pragma unroll
    for (int t = 0; t < 8; ++t) {
#pragma unroll
        for (int v = 0; v < 8; ++v) {
            C[(size_t)(rowBase + v + 8 * lhalf) * GCN_H + 16 * t + llow] = acc[t][v];
        }
    }
}

// ---------------- aggregation ----------------

// agg[i][c] = h[i][c] * dis[i]^2 + bias[c]   (self-loop + conv bias)
__global__ void k_selfloop(const float* __restrict__ h, const float* __restrict__ dis,
                           const float* __restrict__ bias, float* __restrict__ agg,
                           int total) {
    int i = blockIdx.x * blockDim.x + threadIdx.x;
    if (i < total) {
        int row = i >> 7, c = i & (GCN_H - 1);
        float d = dis[row];
        agg[i] = h[i] * d * d + bias[c];
    }
}

// one wave per edge: agg[dst] += h[src] * dis[src]*dis[dst]  (4 floats/lane)
__global__ void __launch_bounds__(256)
k_edge_scatter(const float* __restrict__ h, const int* __restrict__ src,
               const int* __restrict__ dst, const float* __restrict__ dis,
               float* __restrict__ agg, int e) {
    int wave = (int)((blockIdx.x * blockDim.x + threadIdx.x) >> 5);
    if (wave >= e) return;
    int lane = threadIdx.x & 31;
    int s = src[wave], d = dst[wave];
    float coef = dis[s] * dis[d];
    float4 v = ((const float4*)(h + (size_t)s * GCN_H))[lane];
    float* ad = agg + (size_t)d * GCN_H + 4 * lane;
    atomicAdd(ad + 0, v.x * coef);
    atomicAdd(ad + 1, v.y * coef);
    atomicAdd(ad + 2, v.z * coef);
    atomicAdd(ad + 3, v.w * coef);
}

// ---------------- BatchNorm (training stats) + ReLU ----------------

__global__ void __launch_bounds__(128)
k_bn_partial(const float* __restrict__ h, float* __restrict__ sum,
             float* __restrict__ sumsq, int nRows) {
    int c = threadIdx.x;                       // channel
    float s = 0.f, s2 = 0.f;
    for (int r = blockIdx.x; r < nRows; r += gridDim.x) {
        float v = h[(size_t)r * GCN_H + c];
        s += v; s2 += v * v;
    }
    atomicAdd(&sum[c], s);
    atomicAdd(&sumsq[c], s2);
}

__global__ void k_bn_finalize(const float* __restrict__ sum, const float* __restrict__ sumsq,
                              const float* __restrict__ g, const float* __restrict__ be,
                              float* __restrict__ scale, float* __restrict__ shift,
                              float invN) {
    int c = threadIdx.x;
    float mu  = sum[c] * invN;
    float var = sumsq[c] * invN - mu * mu;     // biased variance (jnp.var)
    float sc  = g[c] * rsqrtf(var + 1e-5f);
    scale[c] = sc;
    shift[c] = be[c] - mu * sc;
}

__global__ void k_bn_apply(float* __restrict__ h, const float* __restrict__ scale,
                           const float* __restrict__ shift, int total) {
    int i = blockIdx.x * blockDim.x + threadIdx.x;
    if (i < total) {
        int c = i & (GCN_H - 1);
        float v = h[i] * scale[c] + shift[c];
        h[i] = v > 0.f ? v : 0.f;
    }
}

// ---------------- final linear [N,128] @ [128,1] + bf ----------------
__global__ void __launch_bounds__(256)
k_final(const float* __restrict__ h, const float* __restrict__ Wf,
        const float* __restrict__ bf, float* __restrict__ out, int n) {
    int row = (int)((blockIdx.x * blockDim.x + threadIdx.x) >> 5);
    if (row >= n) return;
    int lane = threadIdx.x & 31;
    float4 v = ((const float4*)(h + (size_t)row * GCN_H))[lane];
    float4 w = ((const float4*)Wf)[lane];
    float p = v.x * w.x + v.y * w.y + v.z * w.z + v.w * w.w;
#pragma unroll
    for (int off = 16; off > 0; off >>= 1) p += __shfl_xor(p, off, 32);
    if (lane == 0) out[row] = p + bf[0];
}

// ---------------- driver ----------------

extern "C" void kernel_launch(void* const* d_in, const int* in_sizes, int n_in,
                              void* d_out, int out_size, void* d_ws, size_t ws_size,
                              hipStream_t stream) {
    (void)in_sizes; (void)n_in; (void)out_size; (void)ws_size;

    const float* x   = (const float*)d_in[0];
    const int*   src = (const int*)  d_in[1];
    const int*   dst = (const int*)  d_in[2];
    const float* W1  = (const float*)d_in[3];
    const float* b1  = (const float*)d_in[4];
    const float* g1  = (const float*)d_in[5];
    const float* be1 = (const float*)d_in[6];
    const float* W2  = (const float*)d_in[7];
    const float* b2  = (const float*)d_in[8];
    const float* g2  = (const float*)d_in[9];
    const float* be2 = (const float*)d_in[10];
    const float* W3  = (const float*)d_in[11];
    const float* b3  = (const float*)d_in[12];
    const float* g3  = (const float*)d_in[13];
    const float* be3 = (const float*)d_in[14];
    const float* Wf  = (const float*)d_in[15];
    const float* bf  = (const float*)d_in[16];
    float* out = (float*)d_out;

    // workspace layout (floats)
    float* fA    = (float*)d_ws;                       // N*H  (post-GEMM h)
    float* fB    = fA + (size_t)GCN_N * GCN_H;         // N*H  (aggregated / activated h)
    float* dis   = fB + (size_t)GCN_N * GCN_H;         // N
    float* sum   = dis + GCN_N;                        // H
    float* sumsq = sum + GCN_H;                        // H
    float* scale = sumsq + GCN_H;                      // H
    float* shift = scale + GCN_H;                      // H

    const int NH = GCN_N * GCN_H;

    // dis = rsqrt(deg + 1): fold +1 by initializing to 1.0
    k_fill  <<<(GCN_N + 255) / 256, 256, 0, stream>>>(dis, 1.0f, GCN_N);
    k_deg   <<<(GCN_E + 255) / 256, 256, 0, stream>>>(dst, dis, GCN_E);
    k_rsqrt <<<(GCN_N + 255) / 256, 256, 0, stream>>>(dis, GCN_N);

    const int gemmBlocks = ((GCN_N / 16) * 32 + 255) / 256;   // one wave per 16 rows
    const int edgeBlocks = (GCN_E * 32 + 255) / 256;          // one wave per edge

    auto layer = [&](const float* in, int K, const float* W, const float* b,
                     const float* g, const float* be) {
        k_gemm_wmma   <<<gemmBlocks, 256, 0, stream>>>(in, W, fA, GCN_N, K);
        k_selfloop    <<<(NH + 255) / 256, 256, 0, stream>>>(fA, dis, b, fB, NH);
        k_edge_scatter<<<edgeBlocks, 256, 0, stream>>>(fA, src, dst, dis, fB, GCN_E);
        k_fill        <<<1, 256, 0, stream>>>(sum, 0.0f, 2 * GCN_H);  // sum+sumsq contiguous
        k_bn_partial  <<<512, 128, 0, stream>>>(fB, sum, sumsq, GCN_N);
        k_bn_finalize <<<1, 128, 0, stream>>>(sum, sumsq, g, be, scale, shift, 1.0f / GCN_N);
        k_bn_apply    <<<(NH + 255) / 256, 256, 0, stream>>>(fB, scale, shift, NH);
    };

    layer(x,  GCN_DIN, W1, b1, g1, be1);
    layer(fB, GCN_H,   W2, b2, g2, be2);
    layer(fB, GCN_H,   W3, b3, g3, be3);

    k_final<<<(GCN_N * 32 + 255) / 256, 256, 0, stream>>>(fB, Wf, bf, out, GCN_N);
}